// DAAAgregationLayer_4784593568033
// MI455X (gfx1250) — compile-verified
//
#include <hip/hip_runtime.h>
#include <hip/hip_bf16.h>

#define DIM 128
#define CHUNK 128   // pairs per wave in the aggregation kernel

typedef float v2f __attribute__((ext_vector_type(2)));
typedef float v8f __attribute__((ext_vector_type(8)));

// ---------------------------------------------------------------------------
// Kernel 0: zero the agg scratch (d_ws is poisoned 0xAA; must re-zero per call)
// ---------------------------------------------------------------------------
__global__ __launch_bounds__(256) void zero_kernel(float4* __restrict__ p, int n4) {
    int idx = blockIdx.x * blockDim.x + threadIdx.x;
    if (idx < n4) p[idx] = make_float4(0.f, 0.f, 0.f, 0.f);
}

// ---------------------------------------------------------------------------
// Kernel 1: segmented pair aggregation.
// One wave (32 lanes) owns CHUNK contiguous pairs; lane l owns dims [4l,4l+4).
// pair_node is sorted, so we accumulate in registers while the segment is
// unchanged and flush with global_atomic_add_f32 only at segment boundaries.
// ---------------------------------------------------------------------------
__global__ __launch_bounds__(256) void pair_agg_kernel(
    const float* __restrict__ features,   // [N, 128]
    const float* __restrict__ dist,       // [N, N]
    const int*   __restrict__ pnode,      // [P] sorted
    const int*   __restrict__ pi,         // [P]
    const int*   __restrict__ pj,         // [P]
    const float* __restrict__ pcos,       // [P]
    float*       __restrict__ agg,        // [N, 128]
    int P, int N)
{
    const int lane = threadIdx.x & 31;
    const int waveInBlk = threadIdx.x >> 5;
    const long wave = (long)blockIdx.x * (blockDim.x >> 5) + waveInBlk;

    long start = wave * CHUNK;
    if (start >= P) return;
    long end = start + CHUNK;
    if (end > P) end = P;

    const int d = lane * 4;
    float ax = 0.f, ay = 0.f, az = 0.f, aw = 0.f;
    int cur = pnode[start];

    for (long p = start; p < end; ++p) {
        int node = pnode[p];
        if (node != cur) {
            float* dst = agg + (long)cur * DIM + d;
            atomicAdd(dst + 0, ax);
            atomicAdd(dst + 1, ay);
            atomicAdd(dst + 2, az);
            atomicAdd(dst + 3, aw);
            ax = ay = az = aw = 0.f;
            cur = node;
        }
        int i = pi[p];
        int j = pj[p];
        float w = pcos[p] * dist[(long)cur * N + i] * dist[(long)cur * N + j];

        const float4 fi = *(const float4*)(features + (long)i * DIM + d);
        const float4 fj = *(const float4*)(features + (long)j * DIM + d);
        ax = fmaf(fi.x + fj.x, w, ax);
        ay = fmaf(fi.y + fj.y, w, ay);
        az = fmaf(fi.z + fj.z, w, az);
        aw = fmaf(fi.w + fj.w, w, aw);
    }
    float* dst = agg + (long)cur * DIM + d;
    atomicAdd(dst + 0, ax);
    atomicAdd(dst + 1, ay);
    atomicAdd(dst + 2, az);
    atomicAdd(dst + 3, aw);
}

// ---------------------------------------------------------------------------
// Kernel 2: out = agg @ W + b using V_WMMA_F32_16X16X4_F32 (full f32 precision).
// One 16x16 output tile per wave; 8 waves/block -> one block = 16-row panel.
// A 16x4 layout : lanes 0-15 = rows M, half-wave selects K {0,1} vs {2,3}.
// B 4x16 layout : N across lanes 0-15, half-wave selects K rows {0,1} vs {2,3}.
// C/D          : 8 VGPRs, lanes 0-15 rows 0-7, lanes 16-31 rows 8-15.
// ---------------------------------------------------------------------------
__global__ __launch_bounds__(256) void gemm_wmma_kernel(
    const float* __restrict__ agg,   // [N, 128]
    const float* __restrict__ Wm,    // [128, 128]
    const float* __restrict__ bias,  // [128]
    float*       __restrict__ out,   // [N, 128]
    int mtiles)                      // N / 16
{
    const int lane  = threadIdx.x & 31;
    const int wave  = threadIdx.x >> 5;
    const int tile  = blockIdx.x * 8 + wave;        // mtiles * 8 tiles total
    const int mtile = tile >> 3;
    const int ntile = tile & 7;
    if (mtile >= mtiles) return;                    // exact fit for N=10000; safety only

    const int m0    = mtile * 16;
    const int n0    = ntile * 16;
    const int lhalf = lane >> 4;                    // 0 or 1
    const int l15   = lane & 15;

    v8f c = {};
    const float* arow = agg + (long)(m0 + l15) * DIM;

    #pragma unroll 4
    for (int k = 0; k < DIM; k += 4) {
        const int ka = k + 2 * lhalf;
        v2f a;
        a.x = arow[ka + 0];
        a.y = arow[ka + 1];
        v2f bmat;
        bmat.x = Wm[(long)(ka + 0) * DIM + n0 + l15];
        bmat.y = Wm[(long)(ka + 1) * DIM + n0 + l15];
        // 8 args: (neg_a, A, neg_b, B, c_mod, C, reuse_a, reuse_b)
        c = __builtin_amdgcn_wmma_f32_16x16x4_f32(
                false, a, false, bmat, (short)0, c, false, false);
    }

    const float bv   = bias[n0 + l15];
    const int   row0 = m0 + lhalf * 8;
    #pragma unroll
    for (int v = 0; v < 8; ++v) {
        out[(long)(row0 + v) * DIM + n0 + l15] = c[v] + bv;
    }
}

// ---------------------------------------------------------------------------
extern "C" void kernel_launch(void* const* d_in, const int* in_sizes, int n_in,
                              void* d_out, int out_size, void* d_ws, size_t ws_size,
                              hipStream_t stream) {
    const float* features = (const float*)d_in[0];
    const float* dist     = (const float*)d_in[1];
    const int*   pnode    = (const int*)  d_in[2];
    const int*   pi       = (const int*)  d_in[3];
    const int*   pj       = (const int*)  d_in[4];
    const float* pcos     = (const float*)d_in[5];
    const float* Wm       = (const float*)d_in[6];
    const float* bias     = (const float*)d_in[7];
    float*       out      = (float*)d_out;

    const int P = in_sizes[2];
    const int N = in_sizes[0] / DIM;

    float* agg = (float*)d_ws;                      // [N, 128] f32 scratch

    // 1) zero agg
    {
        int n4 = (N * DIM) / 4;
        int blocks = (n4 + 255) / 256;
        zero_kernel<<<blocks, 256, 0, stream>>>((float4*)agg, n4);
    }
    // 2) segmented pair aggregation
    {
        long waves  = ((long)P + CHUNK - 1) / CHUNK;
        int  blocks = (int)((waves + 7) / 8);       // 8 waves per 256-thread block
        pair_agg_kernel<<<blocks, 256, 0, stream>>>(features, dist, pnode, pi, pj,
                                                    pcos, agg, P, N);
    }
    // 3) WMMA GEMM: out = agg @ W + b
    {
        int mtiles = N / 16;                        // 625 for N=10000
        int tiles  = mtiles * 8;
        int blocks = (tiles + 7) / 8;
        gemm_wmma_kernel<<<blocks, 256, 0, stream>>>(agg, Wm, bias, out, mtiles);
    }
}